// DeformableConv2d_75788992905230
// MI455X (gfx1250) — compile-verified
//
#include <hip/hip_runtime.h>
#include <hip/hip_bf16.h>
#include <math.h>

// Problem constants (match reference)
#define Bn    8
#define Cn    64
#define Hn    128
#define Wn    128
#define COUTn 64
#define KTAP  9      // 3x3
#define CK    576    // Cn * KTAP
#define TILE_W   32  // pixels per block
#define CHUNK_C  16  // channels per S-chunk
#define CHUNK_K  (CHUNK_C * KTAP)   // 144 rows of S per chunk
#define S_STRIDE 33  // padded row stride (floats) to break bank aliasing

typedef __attribute__((ext_vector_type(2))) float v2f;
typedef __attribute__((ext_vector_type(8))) float v8f;

// LDS partition sizes (in floats)
#define XT_SZ   (Cn * 3 * 34)       // 6528  : x halo tile [c][ry][col]
#define OFFB_SZ (27 * TILE_W)       // 864   : offset/mod conv result [oc][p]
#define TAPW_SZ (KTAP * TILE_W * 4) // 1152  : 4 corner weights (sigmoid folded in)
#define TAPI_SZ (KTAP * TILE_W * 4) // 1152  : 4 clamped corner flat indices
#define S_SZ    (CHUNK_K * S_STRIDE)// 4752  : sample matrix chunk
#define SMEM_FLOATS (XT_SZ + OFFB_SZ + TAPW_SZ + TAPI_SZ + S_SZ)

__global__ __launch_bounds__(256)
void deform_conv2d_wmma_kernel(const float* __restrict__ x,
                               const float* __restrict__ w_main,
                               const float* __restrict__ w_off,
                               const float* __restrict__ w_mod,
                               float* __restrict__ out)
{
    extern __shared__ float smem[];
    float* xt   = smem;                       // [Cn][3][34]
    float* offb = xt + XT_SZ;                 // [27][32]
    float* tapw = offb + OFFB_SZ;             // [(tap*32+p)*4 + corner]
    int*   tapi = (int*)(tapw + TAPW_SZ);     // [(tap*32+p)*4 + corner]
    float* S    = (float*)(tapi + TAPI_SZ);   // [144][S_STRIDE]

    const int w0   = blockIdx.x * TILE_W;
    const int h    = blockIdx.y;
    const int b    = blockIdx.z;
    const int tid  = threadIdx.x;
    const int lane = tid & 31;
    const int wave = tid >> 5;

    const float* xb = x + (size_t)b * Cn * Hn * Wn;

    // ---- Phase 0a: stage x halo tile (rows h-1..h+1, cols w0-1..w0+32) ----
    // In-bounds lanes use the CDNA5 async global->LDS path (ASYNCcnt);
    // padding lanes write zero directly to LDS.
    for (int idx = tid; idx < XT_SZ; idx += 256) {
        int col = idx % 34;
        int t   = idx / 34;
        int ry  = t % 3;
        int c   = t / 3;
        int gy  = h - 1 + ry;
        int gx  = w0 - 1 + col;
        if (gy >= 0 && gy < Hn && gx >= 0 && gx < Wn) {
            const float* gptr = xb + (size_t)c * (Hn * Wn) + gy * Wn + gx;
            // Low 32 bits of an LDS flat pointer are the LDS byte address.
            unsigned ldsoff = (unsigned)(uintptr_t)(xt + idx);
            asm volatile("global_load_async_to_lds_b32 %0, %1, off"
                         :: "v"(ldsoff), "v"(gptr) : "memory");
        } else {
            xt[idx] = 0.0f;
        }
    }
    asm volatile("s_wait_asynccnt 0x0" ::: "memory");
    __syncthreads();

    // ---- Phase 0b: offset (18ch) + modulator (9ch) 3x3 conv over the tile ----
    for (int idx = tid; idx < 27 * TILE_W; idx += 256) {
        int p  = idx & 31;
        int oc = idx >> 5;
        const float* wp = (oc < 18) ? (w_off + (size_t)oc * Cn * 9)
                                    : (w_mod + (size_t)(oc - 18) * Cn * 9);
        float acc = 0.0f;
        for (int c = 0; c < Cn; ++c) {
            const float* xr = xt + c * (3 * 34);
            const float* wc = wp + c * 9;
            #pragma unroll
            for (int ky = 0; ky < 3; ++ky)
                #pragma unroll
                for (int kx = 0; kx < 3; ++kx)
                    acc += wc[ky * 3 + kx] * xr[ky * 34 + p + kx];
        }
        offb[idx] = acc;  // idx == oc*32 + p
    }
    __syncthreads();

    // ---- Phase 0c: per-(tap,pixel) bilinear corner indices + fused weights ----
    for (int idx = tid; idx < KTAP * TILE_W; idx += 256) {
        int p   = idx & 31;
        int tap = idx >> 5;
        float dy = offb[(2 * tap) * TILE_W + p];
        float dx = offb[(2 * tap + 1) * TILE_W + p];
        float mr = offb[(18 + tap) * TILE_W + p];
        float m  = 1.0f / (1.0f + __expf(-mr));      // sigmoid modulator

        float py = dy + (float)(h - 1 + tap / 3);
        float px = dx + (float)(w0 + p - 1 + tap % 3);
        float y0f = floorf(py), x0f = floorf(px);
        float wy = py - y0f,    wx = px - x0f;
        int y0 = (int)y0f, x0i = (int)x0f;
        int y1 = y0 + 1,   x1i = x0i + 1;

        float vy0 = (y0  >= 0 && y0  < Hn) ? 1.0f : 0.0f;
        float vy1 = (y1  >= 0 && y1  < Hn) ? 1.0f : 0.0f;
        float vx0 = (x0i >= 0 && x0i < Wn) ? 1.0f : 0.0f;
        float vx1 = (x1i >= 0 && x1i < Wn) ? 1.0f : 0.0f;

        int iy0 = min(max(y0, 0), Hn - 1);
        int iy1 = min(max(y1, 0), Hn - 1);
        int ix0 = min(max(x0i, 0), Wn - 1);
        int ix1 = min(max(x1i, 0), Wn - 1);

        int e = idx * 4;
        tapi[e + 0] = iy0 * Wn + ix0;
        tapi[e + 1] = iy0 * Wn + ix1;
        tapi[e + 2] = iy1 * Wn + ix0;
        tapi[e + 3] = iy1 * Wn + ix1;
        tapw[e + 0] = (1.0f - wy) * (1.0f - wx) * m * vy0 * vx0;
        tapw[e + 1] = (1.0f - wy) * wx          * m * vy0 * vx1;
        tapw[e + 2] = wy          * (1.0f - wx) * m * vy1 * vx0;
        tapw[e + 3] = wy          * wx          * m * vy1 * vx1;
    }
    __syncthreads();

    // ---- Main: chunked build of S + FP32 WMMA GEMM (64 couts x 32 pixels) ----
    // Wave tiling: 4 cout-groups x 2 pixel-groups of 16x16.
    const int m0   = (wave >> 1) * 16;          // cout base
    const int n0   = (wave & 1) * 16;           // pixel base
    const int ln   = lane & 15;
    const int koff = (lane < 16) ? 0 : 2;       // 16x4 F32 A/B VGPR layout
    const float* aBase = w_main + (size_t)(m0 + ln) * CK + koff;

    v8f acc = {};
    for (int cc = 0; cc < Cn / CHUNK_C; ++cc) {
        const int c0 = cc * CHUNK_C;

        // Gather + modulate into S[kkLocal][p], kk = cLocal*9 + tap
        for (int idx = tid; idx < CHUNK_K * TILE_W; idx += 256) {
            int p   = idx & 31;
            int kkL = idx >> 5;
            int cL  = kkL / 9;
            int tap = kkL - cL * 9;
            const float* xc = xb + (size_t)(c0 + cL) * (Hn * Wn);
            int e = (tap * TILE_W + p) * 4;
            float v = tapw[e + 0] * xc[tapi[e + 0]]
                    + tapw[e + 1] * xc[tapi[e + 1]]
                    + tapw[e + 2] * xc[tapi[e + 2]]
                    + tapw[e + 3] * xc[tapi[e + 3]];
            S[kkL * S_STRIDE + p] = v;
        }
        __syncthreads();

        // 36 WMMA K-steps over this chunk (144 / 4)
        const float* aPtr = aBase + c0 * KTAP;
        #pragma unroll 6
        for (int ks = 0; ks < CHUNK_K / 4; ++ks) {
            int kr = ks * 4;
            v2f a = *(const v2f*)(aPtr + kr);           // L2-resident w_main
            v2f bb;
            bb.x = S[(kr + koff)     * S_STRIDE + n0 + ln];
            bb.y = S[(kr + koff + 1) * S_STRIDE + n0 + ln];
            acc = __builtin_amdgcn_wmma_f32_16x16x4_f32(
                      false, a, false, bb, (short)0, acc, false, false);
        }
        __syncthreads();
    }

    // ---- Store: C/D layout — VGPR r holds M=r (lanes 0-15) / M=r+8 (16-31) ----
    float* ob = out + (size_t)b * COUTn * Hn * Wn + (size_t)h * Wn + w0;
    const int nPix = n0 + (lane & 15);
    const int mAdd = (lane < 16) ? 0 : 8;
    #pragma unroll
    for (int r = 0; r < 8; ++r) {
        int co = m0 + r + mAdd;
        ob[(size_t)co * (Hn * Wn) + nPix] = acc[r];
    }
}

extern "C" void kernel_launch(void* const* d_in, const int* in_sizes, int n_in,
                              void* d_out, int out_size, void* d_ws, size_t ws_size,
                              hipStream_t stream) {
    (void)in_sizes; (void)n_in; (void)out_size; (void)d_ws; (void)ws_size;
    const float* x      = (const float*)d_in[0];
    const float* w_main = (const float*)d_in[1];
    const float* w_off  = (const float*)d_in[2];
    const float* w_mod  = (const float*)d_in[3];
    float* out = (float*)d_out;

    dim3 grid(Wn / TILE_W, Hn, Bn);   // 4 x 128 x 8 = 4096 blocks
    dim3 block(256);                  // 8 wave32 waves
    size_t smemBytes = (size_t)SMEM_FLOATS * sizeof(float);  // ~57.8 KB
    deform_conv2d_wmma_kernel<<<grid, block, smemBytes, stream>>>(
        x, w_main, w_off, w_mod, out);
}